// LightGCN_67946382623132
// MI455X (gfx1250) — compile-verified
//
#include <hip/hip_runtime.h>
#include <cstddef>

// ---------------------------------------------------------------------------
// LightGCN on MI455X (gfx1250, wave32)
//  - SpMM phases: 64 threads/edge, coalesced 256B row gather, hardware f32
//    global atomics for the scatter (L2-resident working set, 192MB L2).
//  - Readout: V_WMMA_F32_16X16X4_F32, one wave -> 16 dot products (diag of
//    16x16 f32 WMMA accumulator), scale 1/16 = mean over 4 layer snapshots
//    on both sides.
// ---------------------------------------------------------------------------

typedef __attribute__((ext_vector_type(2))) float v2f;
typedef __attribute__((ext_vector_type(8))) float v8f;

#define DIMS 64

// ucur/usum <- user_emb ; icur/isum <- item_emb  (single sweep over U*64+I*64)
__global__ void lgcn_init(const float* __restrict__ ue, const float* __restrict__ ie,
                          float* __restrict__ ucur, float* __restrict__ usum,
                          float* __restrict__ icur, float* __restrict__ isum,
                          int un, int in_) {
    int idx = blockIdx.x * blockDim.x + threadIdx.x;
    if (idx < un) {
        float v = ue[idx];
        ucur[idx] = v; usum[idx] = v;
    } else if (idx < un + in_) {
        int j = idx - un;
        float v = ie[j];
        icur[j] = v; isum[j] = v;
    }
}

// ucur += du[row]*user_emb ; icur += dv[row]*item_emb   (du_org recomputed on
// the fly from the ORIGINAL embeddings, per the reference semantics)
__global__ void lgcn_add_deg(const float* __restrict__ ue, const float* __restrict__ du,
                             float* __restrict__ ucur,
                             const float* __restrict__ ie, const float* __restrict__ dv,
                             float* __restrict__ icur, int un, int in_) {
    int idx = blockIdx.x * blockDim.x + threadIdx.x;
    if (idx < un) {
        ucur[idx] = __builtin_fmaf(du[idx >> 6], ue[idx], ucur[idx]);
    } else if (idx < un + in_) {
        int j = idx - un;
        icur[j] = __builtin_fmaf(dv[j >> 6], ie[j], icur[j]);
    }
}

// y[rows[e]][d] += vals[e] * x[cols[e]][d]   -- 64 threads per edge.
// Row gather is a fully coalesced 256B burst; scatter is one f32 hardware
// atomic per lane (random rows -> negligible contention, L2-resident).
__global__ void lgcn_spmm(const int* __restrict__ rows, const int* __restrict__ cols,
                          const float* __restrict__ vals,
                          const float* __restrict__ x, float* __restrict__ y,
                          int nnz) {
    long long t = (long long)blockIdx.x * blockDim.x + threadIdx.x;
    int e = (int)(t >> 6);
    int d = (int)(t & 63);
    if (e >= nnz) return;
    int r = rows[e];
    int c = cols[e];
    float v = vals[e];
    float xv = x[(size_t)c * DIMS + d];
    unsafeAtomicAdd(&y[(size_t)r * DIMS + d], v * xv);
}

// usum += unext ; isum += inext
__global__ void lgcn_accum(float* __restrict__ usum, const float* __restrict__ unext,
                           float* __restrict__ isum, const float* __restrict__ inext,
                           int un, int in_) {
    int idx = blockIdx.x * blockDim.x + threadIdx.x;
    if (idx < un) {
        usum[idx] += unext[idx];
    } else if (idx < un + in_) {
        int j = idx - un;
        isum[j] += inext[j];
    }
}

// gamma[b] = (1/16) * dot(usum[users[b]], isum[items[b]])  over D=64.
// One wave handles 16 batch entries: A = 16 user rows, B = 16 item rows,
// C = A*B^T (16x16) accumulated through 16x V_WMMA_F32_16X16X4_F32 with
// K advanced 4 at a time; gamma is the diagonal of C.
// A-layout (16x4 f32): lanes 0-15 M=lane K={0,1}; lanes 16-31 M=lane-16 K={2,3}.
// B-layout (4x16 f32): lanes 0-15 N=lane K={0,1}; lanes 16-31 N=lane-16 K={2,3}.
// C/D layout: VGPR r, lanes 0-15 -> (M=r, N=lane); lanes 16-31 -> (M=r+8, N=lane-16).
__global__ void lgcn_gamma_wmma(const float* __restrict__ usum, const float* __restrict__ isum,
                                const int* __restrict__ users, const int* __restrict__ items,
                                float* __restrict__ out, int batch) {
    int wave = (int)((blockIdx.x * blockDim.x + threadIdx.x) >> 5);
    int lane = threadIdx.x & 31;
    int base = wave * 16;
    if (base >= batch) return;          // whole-wave predicate: EXEC stays all-1s

    int mn = lane & 15;                 // row (A) / col (B) index handled by this lane
    int khalf = (lane >> 4) << 1;       // lanes 16-31 own K+2,K+3 of each K-quad

    const float* __restrict__ arow = usum + (size_t)users[base + mn] * DIMS;
    const float* __restrict__ brow = isum + (size_t)items[base + mn] * DIMS;

    v8f c = {};
#pragma unroll
    for (int kk = 0; kk < DIMS; kk += 4) {
        v2f a, b;
        a.x = arow[kk + khalf];
        a.y = arow[kk + khalf + 1];
        b.x = brow[kk + khalf];
        b.y = brow[kk + khalf + 1];
        // 8 args: (neg_a, A, neg_b, B, c_mod, C, reuse_a, reuse_b)
        c = __builtin_amdgcn_wmma_f32_16x16x4_f32(
            false, a, false, b, (short)0, c, false, false);
    }

    // Diagonal extraction per the documented C/D VGPR layout.
    // M=0..7  live at (VGPR=M,   lane=M)        -> lanes 0..7
    // M=8..15 live at (VGPR=M-8, lane=M+16)     -> lanes 24..31
    const float scale = 0.0625f;        // (1/4 users-mean) * (1/4 items-mean)
    if (lane < 8) {
        out[base + lane] = c[lane] * scale;
    } else if (lane >= 24) {
        out[base + (lane - 16)] = c[lane - 24] * scale;
    }
}

extern "C" void kernel_launch(void* const* d_in, const int* in_sizes, int n_in,
                              void* d_out, int out_size, void* d_ws, size_t ws_size,
                              hipStream_t stream) {
    const float* user_emb = (const float*)d_in[0];
    const float* item_emb = (const float*)d_in[1];
    const float* du       = (const float*)d_in[2];
    const float* dv       = (const float*)d_in[3];
    const float* ui_vals  = (const float*)d_in[4];
    const float* uu_vals  = (const float*)d_in[5];
    const float* vv_vals  = (const float*)d_in[6];
    const int*   ui_rows  = (const int*)d_in[7];
    const int*   ui_cols  = (const int*)d_in[8];
    const int*   uu_rows  = (const int*)d_in[9];
    const int*   uu_cols  = (const int*)d_in[10];
    const int*   vv_rows  = (const int*)d_in[11];
    const int*   vv_cols  = (const int*)d_in[12];
    const int*   users    = (const int*)d_in[13];
    const int*   items    = (const int*)d_in[14];
    float* gamma = (float*)d_out;

    const int U = in_sizes[2];           // 200000
    const int I = in_sizes[3];           // 100000
    const int nnz_ui = in_sizes[4];      // 2,000,000
    const int nnz_uu = in_sizes[5];      // 1,000,000
    const int nnz_vv = in_sizes[6];      // 1,000,000
    const int batch  = in_sizes[13];     // 4096

    const size_t UN = (size_t)U * DIMS;  // 12.8M floats
    const size_t IN = (size_t)I * DIMS;  // 6.4M floats

    float* ws = (float*)d_ws;
    float* ucur  = ws;
    float* unext = ws + UN;
    float* usum  = ws + 2 * UN;
    float* icur  = ws + 3 * UN;
    float* inext = ws + 3 * UN + IN;
    float* isum  = ws + 3 * UN + 2 * IN;

    const int TPB = 256;
    const unsigned ew_blocks = (unsigned)((UN + IN + TPB - 1) / TPB);
    auto spmm_blocks = [&](int nnz) {
        return (unsigned)(((long long)nnz * 64 + TPB - 1) / TPB);
    };

    lgcn_init<<<ew_blocks, TPB, 0, stream>>>(user_emb, item_emb,
                                             ucur, usum, icur, isum,
                                             (int)UN, (int)IN);

    for (int layer = 0; layer < 3; ++layer) {
        // users_emb += du_org ; items_emb += dv_org
        lgcn_add_deg<<<ew_blocks, TPB, 0, stream>>>(user_emb, du, ucur,
                                                    item_emb, dv, icur,
                                                    (int)UN, (int)IN);
        // new_users = uv + uu
        hipMemsetAsync(unext, 0, UN * sizeof(float), stream);
        lgcn_spmm<<<spmm_blocks(nnz_ui), TPB, 0, stream>>>(ui_rows, ui_cols, ui_vals,
                                                           icur, unext, nnz_ui);
        lgcn_spmm<<<spmm_blocks(nnz_uu), TPB, 0, stream>>>(uu_rows, uu_cols, uu_vals,
                                                           ucur, unext, nnz_uu);
        // new_items = vu(new_users) + vv(items_emb')
        hipMemsetAsync(inext, 0, IN * sizeof(float), stream);
        lgcn_spmm<<<spmm_blocks(nnz_ui), TPB, 0, stream>>>(ui_cols, ui_rows, ui_vals,
                                                           unext, inext, nnz_ui);
        lgcn_spmm<<<spmm_blocks(nnz_vv), TPB, 0, stream>>>(vv_rows, vv_cols, vv_vals,
                                                           icur, inext, nnz_vv);
        // layer snapshot accumulation
        lgcn_accum<<<ew_blocks, TPB, 0, stream>>>(usum, unext, isum, inext,
                                                  (int)UN, (int)IN);
        // swap cur <-> next (deterministic host-side pointer swap)
        float* t;
        t = ucur; ucur = unext; unext = t;
        t = icur; icur = inext; inext = t;
    }

    // gamma readout: 16 pairs per wave via f32 WMMA; 4096/16 = 256 waves.
    const int waves = (batch + 15) / 16;
    const unsigned gblocks = (unsigned)((waves * 32 + TPB - 1) / TPB);
    lgcn_gamma_wmma<<<gblocks, TPB, 0, stream>>>(usum, isum, users, items, gamma, batch);
}